// DecoderRNN_36292473651688
// MI455X (gfx1250) — compile-verified
//
#include <hip/hip_runtime.h>
#include <hip/hip_bf16.h>

typedef __attribute__((ext_vector_type(2))) float v2f;
typedef __attribute__((ext_vector_type(8))) float v8f;

#define ZSZ   128
#define NCOND 40
#define CSZ   100
#define HID   1024
#define INSZ  228          // ZSZ + CSZ
#define FEAT  1252         // HID + INSZ
#define G3    3072         // 3*HID
#define VOCAB 32000
#define TSTEP 512
#define LDK   66           // padded LDS stride (floats)

// ---------------------------------------------------------------- prologue
__global__ void prep_kernel(const float* __restrict__ z, const float* __restrict__ cond,
                            const float* __restrict__ i2h_W, const float* __restrict__ i2h_b,
                            const float* __restrict__ c2h_W, const float* __restrict__ c2h_b,
                            float* __restrict__ de, float* __restrict__ h0) {
  __shared__ float s_de[INSZ];
  int t = threadIdx.x;
  if (t < ZSZ) {
    s_de[t] = z[t];
  } else if (t < INSZ) {
    int j = t - ZSZ;
    float acc = c2h_b[j];
    for (int k = 0; k < NCOND; ++k) acc += c2h_W[j * NCOND + k] * cond[k];
    s_de[t] = acc;
  }
  __syncthreads();
  if (t < INSZ) de[t] = s_de[t];
  for (int i = t; i < HID; i += 256) {
    float acc = i2h_b[i];
    const float* w = i2h_W + (long)i * INSZ;
    for (int k = 0; k < INSZ; ++k) acc += w[k] * s_de[k];
    h0[i] = acc;
  }
}

// ----------------------------------------------- gi for the two token cases
// one wave per row of W_ih; reads the row once, dots against both x_sos/x_unk
__global__ void gi_kernel(const float* __restrict__ embed_W, const float* __restrict__ W_ih,
                          const float* __restrict__ b_ih, const float* __restrict__ de,
                          float* __restrict__ gi_sos, float* __restrict__ gi_unk) {
  int w = threadIdx.x >> 5, lane = threadIdx.x & 31;
  int row = blockIdx.x * 8 + w;                 // 0..3071
  const float* wr = W_ih + (long)row * FEAT;
  float ds = 0.f, du = 0.f;
  for (int k = lane; k < FEAT; k += 32) {
    float xs, xu;
    if (k < HID) {
      xs = fmaxf(embed_W[1 * HID + k], 0.f);    // relu(embed[SOS])
      xu = fmaxf(embed_W[2 * HID + k], 0.f);    // relu(embed[UNK])
    } else {
      xs = xu = de[k - HID];
    }
    float wv = wr[k];
    ds += wv * xs;
    du += wv * xu;
  }
  for (int off = 16; off; off >>= 1) {
    ds += __shfl_xor(ds, off, 32);
    du += __shfl_xor(du, off, 32);
  }
  if (lane == 0) {
    gi_sos[row] = ds + b_ih[row];
    gi_unk[row] = du + b_ih[row];
  }
}

// ------------------------------------------------------- one GRU time-step
// one wave per hidden unit i: computes all three gate dot-products
__global__ void gru_step(const float* __restrict__ W_hh, const float* __restrict__ b_hh,
                         const float* __restrict__ gi, const float* __restrict__ h_prev,
                         float* __restrict__ h_next, float* __restrict__ hs_row) {
  __shared__ float sh[HID];
  int w = threadIdx.x >> 5, lane = threadIdx.x & 31;
  int i = blockIdx.x * 8 + w;                   // 0..1023
  for (int k = threadIdx.x; k < HID; k += 256) sh[k] = h_prev[k];
  __syncthreads();
  const float* wr = W_hh + (long)i * HID;
  const float* wu = W_hh + (long)(HID + i) * HID;
  const float* wn = W_hh + (long)(2 * HID + i) * HID;
  float dr = 0.f, du = 0.f, dn = 0.f;
  for (int k = lane; k < HID; k += 32) {
    float hv = sh[k];
    dr += wr[k] * hv;
    du += wu[k] * hv;
    dn += wn[k] * hv;
  }
  for (int off = 16; off; off >>= 1) {
    dr += __shfl_xor(dr, off, 32);
    du += __shfl_xor(du, off, 32);
    dn += __shfl_xor(dn, off, 32);
  }
  if (lane == 0) {
    float ghr = dr + b_hh[i];
    float ghu = du + b_hh[HID + i];
    float ghn = dn + b_hh[2 * HID + i];
    float r = 1.f / (1.f + __expf(-(gi[i] + ghr)));
    float u = 1.f / (1.f + __expf(-(gi[HID + i] + ghu)));
    float n = tanhf(gi[2 * HID + i] + r * ghn);
    float hv = (1.f - u) * n + u * sh[i];
    h_next[i] = hv;
    hs_row[i] = hv;
  }
}

// ---------------------------------------------- output GEMM, fp32 WMMA 16x16x4
// block = 8 waves; wave owns 16 vocab columns, 8 stacked 16x16 M-tiles.
// 4 M-passes of 128 steps cover T=512; out_W streamed once per pass (L2-resident after pass 0).
__global__ void __launch_bounds__(256) out_gemm(const float* __restrict__ out_W,
                                                const float* __restrict__ out_b,
                                                const float* __restrict__ hs,
                                                const float* __restrict__ de,
                                                float* __restrict__ out) {
  __shared__ float lds[128 * LDK];
  int tid = threadIdx.x;
  int w = tid >> 5, lane = tid & 31;
  int n0 = blockIdx.x * 128 + w * 16;           // wave's vocab tile base
  int nrow = lane & 15;                         // N index in tile; also M-row for A frag
  int koff = (lane >> 4) * 2;                   // 0 or 2 (A/B fragment K split)
  int vcol = n0 + nrow;
  float bias = out_b[vcol];
  const float* brow = out_W + (long)vcol * FEAT;

  for (int mpass = 0; mpass < 4; ++mpass) {
    int mbase = mpass * 128;
    v8f acc[8] = {};
    for (int k0 = 0; k0 < FEAT; k0 += 64) {
      int kc = (FEAT - k0 < 64) ? (FEAT - k0) : 64;   // 64 ... then 36 (mult of 4)
      __syncthreads();
      // stage A: feats[mbase..mbase+127][k0..k0+kc) = hs || de
      for (int r = w; r < 128; r += 8) {
        int m = mbase + r;
        for (int kk = lane; kk < kc; kk += 32) {
          int kg = k0 + kk;
          lds[r * LDK + kk] = (kg < HID) ? hs[(long)m * HID + kg] : de[kg - HID];
        }
      }
      __syncthreads();
      __builtin_prefetch(brow + k0 + 64, 0, 1);       // stream next K chunk of out_W
      for (int kk = 0; kk < kc; kk += 4) {
        v2f bfrag = *(const v2f*)(brow + k0 + kk + koff);   // B[k..k+3][n], 8B aligned
#pragma unroll
        for (int mt = 0; mt < 8; ++mt) {
          v2f afrag = *(const v2f*)&lds[(mt * 16 + nrow) * LDK + kk + koff];
          acc[mt] = __builtin_amdgcn_wmma_f32_16x16x4_f32(
              false, afrag, false, bfrag, (short)0, acc[mt], false, false);
        }
      }
    }
    // D layout: VGPR j -> M=j (lanes 0-15) / M=8+j (lanes 16-31), N = lane&15
    int rowoff = (lane >> 4) * 8;
#pragma unroll
    for (int mt = 0; mt < 8; ++mt) {
      int mrow0 = mbase + mt * 16 + rowoff;
#pragma unroll
      for (int j = 0; j < 8; ++j) {
        out[(long)(mrow0 + j) * VOCAB + vcol] = acc[mt][j] + bias;
      }
    }
  }
}

// ---------------------------------------------------------------- launcher
extern "C" void kernel_launch(void* const* d_in, const int* in_sizes, int n_in,
                              void* d_out, int out_size, void* d_ws, size_t ws_size,
                              hipStream_t stream) {
  const float* z       = (const float*)d_in[0];
  const float* cond    = (const float*)d_in[1];
  // d_in[2] = inputs (unused: word_dropout=1.0 makes tokens deterministic)
  // d_in[3] = temperature (unused: logits returned un-scaled by reference)
  const float* embed_W = (const float*)d_in[4];
  const float* W_ih    = (const float*)d_in[5];
  const float* W_hh    = (const float*)d_in[6];
  const float* b_ih    = (const float*)d_in[7];
  const float* b_hh    = (const float*)d_in[8];
  const float* i2h_W   = (const float*)d_in[9];
  const float* i2h_b   = (const float*)d_in[10];
  const float* c2h_W   = (const float*)d_in[11];
  const float* c2h_b   = (const float*)d_in[12];
  const float* out_W   = (const float*)d_in[13];
  const float* out_b   = (const float*)d_in[14];
  float* out = (float*)d_out;

  float* wsf    = (float*)d_ws;
  float* de     = wsf;                       // 228 (pad to 256)
  float* gi_sos = wsf + 256;                 // 3072
  float* gi_unk = wsf + 256 + G3;            // 3072
  float* hbuf   = wsf + 256 + 2 * G3;        // 2 x 1024 ping-pong
  float* hs     = hbuf + 2 * HID;            // 512 x 1024

  prep_kernel<<<1, 256, 0, stream>>>(z, cond, i2h_W, i2h_b, c2h_W, c2h_b, de, hbuf);
  gi_kernel<<<G3 / 8, 256, 0, stream>>>(embed_W, W_ih, b_ih, de, gi_sos, gi_unk);

  for (int t = 0; t < TSTEP; ++t) {
    const float* gi = (t == 0) ? gi_sos : gi_unk;
    const float* hp = hbuf + (t & 1) * HID;
    float*       hn = hbuf + ((t + 1) & 1) * HID;
    gru_step<<<HID / 8, 256, 0, stream>>>(W_hh, b_hh, gi, hp, hn, hs + (long)t * HID);
  }

  out_gemm<<<VOCAB / 128, 256, 0, stream>>>(out_W, out_b, hs, de, out);
}